// N3AggregationBase_71511205478647
// MI455X (gfx1250) — compile-verified
//
#include <hip/hip_runtime.h>
#include <hip/hip_bf16.h>

// ---------------------------------------------------------------------------
// Problem constants (from reference)
// ---------------------------------------------------------------------------
#define KNN   7        // neighbor volumes
#define PSZ   8        // patch size
#define CCH   8        // channels per patch; F = CCH*PSZ*PSZ = 512
#define HIMG  512
#define WIMG  512
#define NPTS  16384
#define QNUM  4096
#define ONUM  64
#define EDIM  256
#define FDIM  512      // CCH*PSZ*PSZ
#define NWCOL 505      // WIMG - PSZ + 1
#define KC    (KNN*CCH)          // 56 output channels
#define HW    (HIMG*WIMG)        // 262144

#define RPS     8      // rows per async stage (8 x 512 f32 = 16 KB)
#define NSTAGE  (ONUM / RPS)     // 8 stages
#define SXPITCH 520    // sX row stride (floats): bank = (8*row + f) % 64
                       // -> half-waves (rows r / r+2, bank offset 16) disjoint

typedef __attribute__((ext_vector_type(2))) float v2f;
typedef __attribute__((ext_vector_type(8))) float v8f;

// ---------------------------------------------------------------------------
// Zero fill
// ---------------------------------------------------------------------------
__global__ void zero_f32(float* __restrict__ p, int n) {
  int i = blockIdx.x * blockDim.x + threadIdx.x;
  if (i < n) p[i] = 0.0f;
}

// ---------------------------------------------------------------------------
// Search: logits_base[q,o] = 2*dot(ye[q], xe[qinds[q,o]]) - ||xe_g||^2.
// (-||ye[q]||^2 is constant over o and cancels in every softmax -> dropped.)
// One thread per (q,o); float4 loads; xe/ye are L2-resident (20 MB total).
// ---------------------------------------------------------------------------
__global__ void dist_kernel(const float* __restrict__ xe,
                            const float* __restrict__ ye,
                            const int* __restrict__ qinds,
                            float* __restrict__ dists) {
  int gid = blockIdx.x * blockDim.x + threadIdx.x;   // QNUM*ONUM threads
  int q = gid >> 6;
  int idx = qinds[gid];
  const float4* xr = (const float4*)(xe + (size_t)idx * EDIM);
  const float4* yr = (const float4*)(ye + (size_t)q * EDIM);
  float dot = 0.f, sx = 0.f;
#pragma unroll 8
  for (int e = 0; e < EDIM / 4; ++e) {
    float4 a = xr[e];
    float4 b = yr[e];
    dot += a.x * b.x + a.y * b.y + a.z * b.z + a.w * b.w;
    sx  += a.x * a.x + a.y * a.y + a.z * a.z + a.w * a.w;
  }
  dists[gid] = 2.0f * dot - sx;
}

// ---------------------------------------------------------------------------
// Temperature: mean of 8x8 log_temp patch at qindex, then exp().
// (TEMP_BIAS = 1.0 -> log(TEMP_BIAS) = 0.)
// ---------------------------------------------------------------------------
__global__ void temp_kernel(const float* __restrict__ log_temp,
                            const int* __restrict__ qindex,
                            float* __restrict__ temp) {
  int q = blockIdx.x * blockDim.x + threadIdx.x;
  if (q >= QNUM) return;
  int qi = qindex[q];
  int row = qi / NWCOL, col = qi % NWCOL;
  float s = 0.f;
#pragma unroll
  for (int pi = 0; pi < PSZ; ++pi)
#pragma unroll
    for (int pj = 0; pj < PSZ; ++pj)
      s += log_temp[(row + pi) * WIMG + col + pj];
  temp[q] = __expf(s * (1.0f / (PSZ * PSZ)));
}

// ---------------------------------------------------------------------------
// NeuralNearestNeighbors weights: K=7 rounds of masked softmax over 64
// candidates. One wave32 per query; each lane owns o and o+32; cross-lane
// max/sum via __shfl_xor (wave32!). Output layout W[q][k][o] so it is
// directly the A matrix (M=k rows, K-dim=o) of the wpsum GEMM.
// ---------------------------------------------------------------------------
__global__ void nnn_weights_kernel(const float* __restrict__ dists,
                                   const float* __restrict__ temp,
                                   float* __restrict__ Wmat) {
  int q = blockIdx.x;
  int lane = threadIdx.x;                 // 0..31
  float invT = 1.0f / temp[q];
  float l0 = dists[q * ONUM + lane] * invT;
  float l1 = dists[q * ONUM + lane + 32] * invT;
  float M0 = 0.f, M1 = 0.f;
  for (int k = 0; k < KNN; ++k) {
    float a0 = l0 + M0, a1 = l1 + M1;
    float m = fmaxf(a0, a1);
#pragma unroll
    for (int d = 16; d > 0; d >>= 1) m = fmaxf(m, __shfl_xor(m, d, 32));
    float e0 = __expf(a0 - m), e1 = __expf(a1 - m);
    float s = e0 + e1;
#pragma unroll
    for (int d = 16; d > 0; d >>= 1) s += __shfl_xor(s, d, 32);
    float w0 = e0 / s, w1 = e1 / s;
    Wmat[(q * KNN + k) * ONUM + lane]      = w0;
    Wmat[(q * KNN + k) * ONUM + lane + 32] = w1;
    M0 += __logf(fmaxf(1.0f - w0, 1e-6f));   // soft exclusion mask
    M1 += __logf(fmaxf(1.0f - w1, 1e-6f));
  }
}

// ---------------------------------------------------------------------------
// wpsum + fold: per query q, z = W^T (7x64, padded to 16) x x_g (64x512)
// with V_WMMA_F32_16X16X4_F32, then scatter-add (fold) the 56-channel 8x8
// patch into img with global_atomic_add_f32.
//
// One block (4 waves / 128 threads) per query. Fully software-pipelined:
//  * x_g staged through a DOUBLE-BUFFERED LDS tile (8 gathered rows = 16 KB
//    per stage, 8 stages) with gfx1250 async copies:
//    global_load_async_to_lds_b128, then a PARTIAL wait
//    (s_wait_asynccnt 0x8: async loads complete in order, so <=8 outstanding
//    means the current stage is resident) so stage s+1's DMA overlaps
//    stage s's WMMAs.
//  * Stage loop fully unrolled -> afrag[] indices are compile-time constants
//    (no v_movrels / m0 indexing).
//  * All 8 accumulators (one per f-tile of the wave) live in VGPRs across
//    stages -> 8 independent WMMAs between dependent ones.
//
// A-fragment layout (ISA 16x4 f32 A): lanes 0-15 -> row M=lane, K=0,1;
// lanes 16-31 -> row M=lane-16, K=2,3. B (4x16): lanes 0-15 -> col N=lane,
// K=0,1; lanes 16-31 -> col N=lane-16, K=2,3. C/D: VGPR v, lanes 0-15 hold
// row M=v (rows 0..6 are the 7 k-slices; rows 7..15 stay zero).
// ---------------------------------------------------------------------------
__global__ void wpsum_fold_kernel(const float* __restrict__ x,
                                  const float* __restrict__ Wmat,
                                  const int* __restrict__ qinds,
                                  const int* __restrict__ qindex,
                                  float* __restrict__ img) {
  __shared__ float sW[16 * 65];              // 16x64 W tile, stride 65
  __shared__ float sX[2][RPS * SXPITCH];     // double-buffered x-row stage
  __shared__ int   sIdx[ONUM];

  int q   = blockIdx.x;
  int tid = threadIdx.x;               // 0..127

  // Stage W[q] (7x64) into LDS, zero-padding rows 7..15.
  for (int i = tid; i < 16 * 64; i += 128) {
    int r = i >> 6, c = i & 63;
    sW[r * 65 + c] = (r < KNN) ? Wmat[(q * KNN + r) * ONUM + c] : 0.0f;
  }
  if (tid < ONUM) {
    int id = qinds[q * ONUM + tid];
    sIdx[tid] = id;
    __builtin_prefetch(x + (size_t)id * FDIM, 0, 0);   // global_prefetch_b8
  }
  __syncthreads();

  int qi = qindex[q];
  int prow = qi / NWCOL, pcol = qi % NWCOL;

  int wave = tid >> 5;
  int lane = tid & 31;
  bool hi  = lane >= 16;
  int  lr  = lane & 15;

  // Preload all 16 A fragments (whole 16x64 W tile) into VGPRs.
  v2f afrag[16];
#pragma unroll
  for (int ks = 0; ks < 16; ++ks) {
    int kb = ks * 4 + (hi ? 2 : 0);
    v2f a;
    a.x = sW[lr * 65 + kb];
    a.y = sW[lr * 65 + kb + 1];
    afrag[ks] = a;
  }

  v8f acc[8];
#pragma unroll
  for (int t = 0; t < 8; ++t) acc[t] = (v8f){};

  // Issue stage s (8 gathered rows, 2 KB each) into buffer buf.
  // Per row all 128 threads copy 16 B -> fully coalesced async b128 copies.
  // Generic LDS pointer low 32 bits == LDS byte offset (ISA 10.2).
  auto issue_stage = [&](int s, int buf) {
#pragma unroll
    for (int i = 0; i < RPS; ++i) {
      const float* gp = x + (size_t)sIdx[s * RPS + i] * FDIM + tid * 4;
      unsigned lp = (unsigned)(size_t)&sX[buf][i * SXPITCH + tid * 4];
      asm volatile("global_load_async_to_lds_b128 %0, %1, off"
                   :: "v"(lp), "v"(gp) : "memory");
    }
  };

  issue_stage(0, 0);

#pragma unroll
  for (int s = 0; s < NSTAGE; ++s) {
    int cur = s & 1;
    if (s + 1 < NSTAGE) {
      issue_stage(s + 1, cur ^ 1);                       // overlap next DMA
      asm volatile("s_wait_asynccnt 0x8" ::: "memory");  // stage s resident
    } else {
      asm volatile("s_wait_asynccnt 0x0" ::: "memory");
    }
    __syncthreads();   // stage s visible to all waves

#pragma unroll
    for (int kl = 0; kl < 2; ++kl) {
      int ks = s * 2 + kl;                 // compile-time constant
      int rb = kl * 4 + (hi ? 2 : 0);
#pragma unroll
      for (int t = 0; t < 8; ++t) {
        int f = (wave * 8 + t) * 16 + lr;
        v2f b;
        b.x = sX[cur][rb * SXPITCH + f];
        b.y = sX[cur][(rb + 1) * SXPITCH + f];
        acc[t] = __builtin_amdgcn_wmma_f32_16x16x4_f32(
            /*neg_a=*/false, afrag[ks], /*neg_b=*/false, b,
            /*c_mod=*/(short)0, acc[t], /*reuse_a=*/false, /*reuse_b=*/false);
      }
    }
    __syncthreads();   // all reads of buf[cur] done before stage s+2 reuses it
  }

  // Fold: lanes 0-15 hold rows 0..7 of C/D; rows 0..6 are the 7 k-slices.
  if (!hi) {
#pragma unroll
    for (int t = 0; t < 8; ++t) {
      int f = (wave * 8 + t) * 16 + lr;
      int c  = f >> 6;                         // channel within patch-feature
      int p  = f & 63;                         // pixel within 8x8 patch
      int pix = (prow + (p >> 3)) * WIMG + pcol + (p & 7);
#pragma unroll
      for (int v = 0; v < KNN; ++v)
        atomicAdd(&img[(size_t)(v * CCH + c) * HW + pix], acc[t][v]);
    }
  }
}

// ---------------------------------------------------------------------------
// Ones-fold: per-pixel hit count (shared by all 56 channels).
// ---------------------------------------------------------------------------
__global__ void count_kernel(const int* __restrict__ qindex,
                             float* __restrict__ count) {
  int q = blockIdx.x;
  int t = threadIdx.x;                         // 0..63 = pixel within patch
  int qi = qindex[q];
  int prow = qi / NWCOL, pcol = qi % NWCOL;
  int pix = (prow + (t >> 3)) * WIMG + pcol + (t & 7);
  atomicAdd(&count[pix], 1.0f);
}

// wimg[kc, p] = count[p] broadcast over the 56 channels.
__global__ void wimg_kernel(const float* __restrict__ count,
                            float* __restrict__ wimg) {
  int i = blockIdx.x * blockDim.x + threadIdx.x;   // KC*HW threads
  wimg[i] = count[i & (HW - 1)];
}

// ---------------------------------------------------------------------------
// Launcher
// ---------------------------------------------------------------------------
extern "C" void kernel_launch(void* const* d_in, const int* in_sizes, int n_in,
                              void* d_out, int out_size, void* d_ws, size_t ws_size,
                              hipStream_t stream) {
  (void)in_sizes; (void)n_in; (void)out_size; (void)ws_size;

  const float* x        = (const float*)d_in[0];  // (1,N,F)
  const float* xe       = (const float*)d_in[1];  // (1,N,E)
  const float* ye       = (const float*)d_in[2];  // (1,Q,E)
  const float* log_temp = (const float*)d_in[3];  // (1,1,H,W)
  const int*   qinds    = (const int*)d_in[4];    // (1,Q,O)
  const int*   qindex   = (const int*)d_in[5];    // (Q,)

  float* img  = (float*)d_out;                    // (KC,H,W)
  float* wimg = img + (size_t)KC * HW;            // (KC,H,W)

  // Workspace layout (floats): dists | temp | W | count  (~9.5 MB total)
  float* ws    = (float*)d_ws;
  float* dists = ws;                              // Q*O
  float* temp  = dists + QNUM * ONUM;             // Q
  float* Wmat  = temp + QNUM;                     // Q*KNN*ONUM
  float* count = Wmat + QNUM * KNN * ONUM;        // H*W

  const int nimg = KC * HW;                       // 14,680,064

  zero_f32<<<(nimg + 255) / 256, 256, 0, stream>>>(img, nimg);
  zero_f32<<<(HW + 255) / 256, 256, 0, stream>>>(count, HW);

  dist_kernel<<<(QNUM * ONUM) / 256, 256, 0, stream>>>(xe, ye, qinds, dists);
  temp_kernel<<<QNUM / 256, 256, 0, stream>>>(log_temp, qindex, temp);
  nnn_weights_kernel<<<QNUM, 32, 0, stream>>>(dists, temp, Wmat);
  wpsum_fold_kernel<<<QNUM, 128, 0, stream>>>(x, Wmat, qinds, qindex, img);
  count_kernel<<<QNUM, 64, 0, stream>>>(qindex, count);
  wimg_kernel<<<nimg / 256, 256, 0, stream>>>(count, wimg);
}